// SuperLoss_35880156791219
// MI455X (gfx1250) — compile-verified
//
#include <hip/hip_runtime.h>
#include <math.h>

// SuperLoss for MI455X (gfx1250, wave32).
//   out[0]      = mean(loss)
//   out[1..N]   = cl_score = l_i - log(10)
//   out[N+1]    = tau = log(10)
//
// Roofline: ~256MB traffic -> ~11us @ 23.3 TB/s; the Halley loop makes this
// VALU-bound unless divisions are cheap. This version uses a single-reciprocal
// Halley form with hardware v_rcp_f32 (1 ULP; Halley is self-correcting), and
// v_log_f32/v_exp_f32 fast paths, cutting ~2x VALU vs the IEEE-div version.
// Input (128MB) fits the 192MB L2 -> RT loads; cl_score stores non-temporal so
// they don't evict it. Mean reduction: deterministic two-pass, final sum via
// V_WMMA_F32_16X16X4_F32 (ones-matrix column sums).

#define TAU_F      2.302585092994046f      // log(10)
#define NEG_INV_E  (-0.36787944117144233f) // -1/e
#define E_F        2.7182818284590452f

typedef float v2f __attribute__((ext_vector_type(2)));
typedef float v8f __attribute__((ext_vector_type(8)));

__device__ __forceinline__ float fast_rcp(float x) {
    return __builtin_amdgcn_rcpf(x);       // v_rcp_f32, ~1 ULP
}

__device__ __forceinline__ float lambertw_halley(float y) {
    // Branch-point series init near y=-1/e, log init for y>=0.9 (as reference).
    float p  = sqrtf(fmaxf(fmaf(2.0f * E_F, y, 2.0f), 0.0f));
    float wb = fmaf(p, fmaf(p, fmaf(p, 11.0f / 72.0f, -1.0f / 3.0f), 1.0f), -1.0f);
    float wl = __logf(1.0f + y);           // y>=0.9 here -> no cancellation
    float w  = (y < 0.9f) ? wb : wl;
#pragma unroll
    for (int it = 0; it < 5; ++it) {       // cubic convergence: 5 iters >= fp32 eps
        float ew = __expf(w);              // v_exp_f32 path
        float f  = fmaf(w, ew, -y);
        float t  = w + 1.0f;
        // Halley, numerator & denominator scaled by 2(w+1):
        //   w -= 2*f*t / (2*ew*t^2 - (w+2)*f)
        float tt = t * t;
        float D  = fmaf(ew + ew, tt, -(w + 2.0f) * f);
        D        = (fabsf(D) < 1e-12f) ? 1e-12f : D;
        w        = fmaf(-(f * (t + t)), fast_rcp(D), w);
    }
    return w;
}

__device__ __forceinline__ float elem_loss(float li, float* cl_out) {
    float cl = li - TAU_F;
    *cl_out  = cl;
    // y_ = 0.5*max(-2/e, cl/0.5) == max(-1/e, cl)
    float y     = fmaxf(NEG_INV_E, cl);
    float w     = lambertw_halley(y);
    float sigma = __expf(-w);
    // (l - tau)*sigma + lam*w^2, lam = 0.5
    return fmaf(cl, sigma, 0.5f * w * w);
}

__global__ void superloss_main(const float* __restrict__ l,
                               float* __restrict__ out_cl,      // d_out + 1 (4B aligned)
                               float* __restrict__ partials,
                               int n) {
    const int tid      = blockIdx.x * blockDim.x + threadIdx.x;
    const int nthreads = gridDim.x * blockDim.x;
    const int nvec     = n >> 2;

    const float4* __restrict__ l4 = (const float4*)l;
    float acc = 0.0f;

    for (int i = tid; i < nvec; i += nthreads) {
        float4 v = l4[i];                     // global_load_b128, coalesced
        float c0, c1, c2, c3;
        float s0 = elem_loss(v.x, &c0);
        float s1 = elem_loss(v.y, &c1);
        float s2 = elem_loss(v.z, &c2);
        float s3 = elem_loss(v.w, &c3);
        // out_cl is only 4B-aligned (d_out+1): scalar NT stores, one contiguous
        // 512B span per wave. NT keeps L2 free for the (resident) input.
        const int j = i << 2;
        __builtin_nontemporal_store(c0, out_cl + j + 0);
        __builtin_nontemporal_store(c1, out_cl + j + 1);
        __builtin_nontemporal_store(c2, out_cl + j + 2);
        __builtin_nontemporal_store(c3, out_cl + j + 3);
        acc += (s0 + s1) + (s2 + s3);
    }
    // scalar tail (N=2^25 divisible by 4; kept for safety)
    for (int i = (nvec << 2) + tid; i < n; i += nthreads) {
        float c;
        float s = elem_loss(l[i], &c);
        __builtin_nontemporal_store(c, out_cl + i);
        acc += s;
    }

    __shared__ float sm[256];
    sm[threadIdx.x] = acc;
    __syncthreads();
#pragma unroll
    for (int s = 128; s > 0; s >>= 1) {
        if ((int)threadIdx.x < s) sm[threadIdx.x] += sm[threadIdx.x + s];
        __syncthreads();
    }
    if (threadIdx.x == 0) partials[blockIdx.x] = sm[0];
}

// Single-wave finalize: deterministic sum of block partials; final add via
// V_WMMA_F32_16X16X4_F32 with all-ones A. With A==1, D[m][n] = sum_k B[k][n],
// so all 64 B slots (2 VGPRs x 32 lanes) are counted exactly once; lane
// partials go in B vgpr0, zeros in vgpr1; sum(D row 0, lanes 0..15) = total.
// EXEC all ones (32-thread block, reconverged) as WMMA requires.
__global__ void superloss_finalize(const float* __restrict__ partials, int nparts,
                                   float* __restrict__ out, int n) {
    const int lane = threadIdx.x;   // blockDim.x == 32 (one wave32)
    float s = 0.0f;
    for (int i = lane; i < nparts; i += 32) s += partials[i];

    v2f a; a[0] = 1.0f; a[1] = 1.0f;     // ones A-matrix (16x4)
    v2f b; b[0] = s;    b[1] = 0.0f;     // lane partials into B (4x16)
    v8f c = {};
    // (neg_a, A, neg_b, B, c_mod, C, reuse_a, reuse_b)
    c = __builtin_amdgcn_wmma_f32_16x16x4_f32(false, a, false, b,
                                              (short)0, c, false, false);
    float colsum = c[0];                 // D row 0: lanes 0..15 hold N=0..15
#pragma unroll
    for (int off = 8; off > 0; off >>= 1)
        colsum += __shfl_xor(colsum, off, 32);  // reduce lanes 0..15

    if (lane == 0) {
        out[0]     = colsum / (float)n;  // loss_mean
        out[n + 1] = TAU_F;              // tau
    }
}

extern "C" void kernel_launch(void* const* d_in, const int* in_sizes, int n_in,
                              void* d_out, int out_size, void* d_ws, size_t ws_size,
                              hipStream_t stream) {
    const int    n        = in_sizes[0];
    const float* l        = (const float*)d_in[0];
    float*       out      = (float*)d_out;
    float*       partials = (float*)d_ws;

    const int BLOCK = 256;                         // 8 waves per workgroup
    int grid = 2048;                               // 8KB of partials in d_ws
    if (ws_size < (size_t)grid * sizeof(float))
        grid = (int)(ws_size / sizeof(float));
    if (grid < 1) grid = 1;

    superloss_main<<<grid, BLOCK, 0, stream>>>(l, out + 1, partials, n);
    superloss_finalize<<<1, 32, 0, stream>>>(partials, grid, out, n);
}